// Distance_60163901883032
// MI455X (gfx1250) — compile-verified
//
#include <hip/hip_runtime.h>
#include <hip/hip_bf16.h>
#include <math.h>

// Problem geometry (fixed by the reference): pred/target are [8,192,192] f32.
#define Bn   8
#define Hn   192
#define Wn   192
#define PLANE (Hn * Wn)          // 36864
#define NTOT  (Bn * PLANE)       // 294912
#define SENT  1.0e6f             // sentinel dist; SENT^2 == 1e12 == reference INF exactly

typedef __attribute__((ext_vector_type(2))) float v2f;
typedef __attribute__((ext_vector_type(8))) float v8f;

__device__ __forceinline__ float sigmoidf_(float x) {
    return 1.0f / (1.0f + expf(-x));
}

// ---------------------------------------------------------------------------
// Kernel 0: zero per-launch state (has_fg flags + scalar accumulator).
// ---------------------------------------------------------------------------
__global__ void k_init(int* __restrict__ flags, float* __restrict__ acc) {
    int t = threadIdx.x;
    if (t < 16) flags[t] = 0;
    if (t == 16) *acc = 0.0f;
}

// ---------------------------------------------------------------------------
// Kernel 1: DT pass along H (axis 1). f ∈ {0, INF} so the squared DT is just
// (distance to nearest zero)^2 -> two O(n) scans per column, register only.
// blockIdx.x = field (0:pred_fg 1:pred_bg 2:targ_fg 3:targ_bg), blockIdx.y = b,
// threadIdx.x = w (coalesced across the wave at each h).
// Also records per-batch "mask has any foreground" flags.
// ---------------------------------------------------------------------------
__global__ void k_pass1(const float* __restrict__ pred,
                        const float* __restrict__ target,
                        float* __restrict__ g,
                        int* __restrict__ flags) {
    const int f = blockIdx.x;      // 0..3
    const int b = blockIdx.y;      // 0..7
    const int w = threadIdx.x;     // 0..191
    const float* col = ((f < 2) ? pred : target) + (size_t)b * PLANE + w;
    float* gcol = g + (size_t)f * NTOT + (size_t)b * PLANE + w;
    const int inv = f & 1;

    // forward scan: store (unsquared) distance to nearest zero above.
    int last = -1;
    int anyraw = 0;
    for (int h = 0; h < Hn; ++h) {
        float v = col[(size_t)h * Wn];
        int raw = (f < 2) ? (sigmoidf_(v) > 0.5f) : (v > 0.5f);
        anyraw |= raw;
        int m = inv ? !raw : raw;      // mask for this field
        if (!m) last = h;
        float df = (last >= 0) ? (float)(h - last) : SENT;
        gcol[(size_t)h * Wn] = df;     // df == 0  <=>  pixel is a "zero" of f
    }
    if ((f == 0 || f == 2) && anyraw) atomicOr(&flags[b * 2 + (f >> 1)], 1);

    // backward scan: combine with distance to nearest zero below, square.
    int next = -1;
    for (int h = Hn - 1; h >= 0; --h) {
        float gv = gcol[(size_t)h * Wn];
        if (gv == 0.0f) next = h;
        float db = (next >= 0) ? (float)(next - h) : SENT;
        float d = fminf(gv, db);
        gcol[(size_t)h * Wn] = d * d;
    }
}

// ---------------------------------------------------------------------------
// Kernel 2: DT pass along W (general min-plus, brute force over q with LDS
// broadcast reads) fused with the loss epilogue and an exact-f32 WMMA-based
// wave reduction + one atomic per block.
// Block = one (b,h) row, 192 threads (p = column), 6 wave32 waves.
// ---------------------------------------------------------------------------
__global__ void k_pass2(const float* __restrict__ pred,
                        const float* __restrict__ target,
                        const float* __restrict__ g,
                        const int* __restrict__ flags,
                        float* __restrict__ acc) {
    __shared__ float s0[Wn], s1[Wn], s2[Wn], s3[Wn];
    __shared__ float wsum[6];

    const int h = blockIdx.x;
    const int b = blockIdx.y;
    const int p = threadIdx.x;
    const size_t row = (size_t)b * PLANE + (size_t)h * Wn;

    s0[p] = g[0 * (size_t)NTOT + row + p];
    s1[p] = g[1 * (size_t)NTOT + row + p];
    s2[p] = g[2 * (size_t)NTOT + row + p];
    s3[p] = g[3 * (size_t)NTOT + row + p];
    __syncthreads();

    // min-plus DT: same q for all lanes -> LDS broadcast, conflict free.
    float d0 = 3.0e38f, d1 = 3.0e38f, d2 = 3.0e38f, d3 = 3.0e38f;
#pragma unroll 4
    for (int q = 0; q < Wn; ++q) {
        float dq = (float)(p - q);
        float c = dq * dq;
        d0 = fminf(d0, s0[q] + c);
        d1 = fminf(d1, s1[q] + c);
        d2 = fminf(d2, s2[q] + c);
        d3 = fminf(d3, s3[q] + c);
    }

    float pd = flags[b * 2 + 0] ? (sqrtf(d0) + sqrtf(d1)) : 0.0f;
    float td = flags[b * 2 + 1] ? (sqrtf(d2) + sqrtf(d3)) : 0.0f;

    float sp = sigmoidf_(pred[row + p]);
    float tv = target[row + p];
    float err = (sp - tv) * (sp - tv);
    float num = pd * pd + td * td;
    float den = pd + td;
    float val = err * (num / (den * den));

    // --- exact f32 wave32 reduction via V_WMMA_F32_16X16X4_F32 ------------
    // A (16x4): lane L<16 holds A[L][0]=val, A[L][1]=0; lane L>=16 holds
    // A[L-16][2]=val, A[L-16][3]=0.  B = ones(4x16).  D[m][n] = rowsum[m]
    // = val(lane m) + val(lane m+16), replicated over n.
    v2f a;  a.x = val;  a.y = 0.0f;
    v2f bb; bb.x = 1.0f; bb.y = 1.0f;
    v8f c = {};
    c = __builtin_amdgcn_wmma_f32_16x16x4_f32(
        /*neg_a=*/false, a, /*neg_b=*/false, bb,
        /*c_mod=*/(short)0, c, /*reuse_a=*/false, /*reuse_b=*/false);
    // lanes 0-15 hold D rows m=0..7, lanes 16-31 hold m=8..15: per-lane sum
    // of the 8 accumulators gives a half-wave sum; xor-16 combines halves.
    float s = c[0] + c[1] + c[2] + c[3] + c[4] + c[5] + c[6] + c[7];
    float wtot = s + __shfl_xor(s, 16, 32);

    int lane = threadIdx.x & 31;
    int wid  = threadIdx.x >> 5;
    if (lane == 0) wsum[wid] = wtot;
    __syncthreads();
    if (threadIdx.x == 0) {
        float bs = wsum[0] + wsum[1] + wsum[2] + wsum[3] + wsum[4] + wsum[5];
        atomicAdd(acc, bs);
    }
}

// ---------------------------------------------------------------------------
// Kernel 3: finalize the mean.
// ---------------------------------------------------------------------------
__global__ void k_final(const float* __restrict__ acc, float* __restrict__ out) {
    out[0] = acc[0] * (1.0f / (float)NTOT);
}

extern "C" void kernel_launch(void* const* d_in, const int* in_sizes, int n_in,
                              void* d_out, int out_size, void* d_ws, size_t ws_size,
                              hipStream_t stream) {
    (void)in_sizes; (void)n_in; (void)out_size; (void)ws_size;
    const float* pred   = (const float*)d_in[0];
    const float* target = (const float*)d_in[1];

    // workspace layout: 4 DT intermediate planes, then flags, then accumulator
    float* g     = (float*)d_ws;                       // 4 * 294912 f32
    int*   flags = (int*)(g + 4 * (size_t)NTOT);       // 16 ints
    float* acc   = (float*)(flags + 16);               // 1 f32

    k_init <<<1, 32, 0, stream>>>(flags, acc);
    k_pass1<<<dim3(4, Bn), Wn, 0, stream>>>(pred, target, g, flags);
    k_pass2<<<dim3(Hn, Bn), Wn, 0, stream>>>(pred, target, g, flags, acc);
    k_final<<<1, 1, 0, stream>>>(acc, (float*)d_out);
}